// KalmanFilter_52072183497423
// MI455X (gfx1250) — compile-verified
//
#include <hip/hip_runtime.h>
#include <hip/hip_bf16.h>

// Kalman filter, B=2048 batches, T=64 steps, STATE=32, OBS=16.
// One wave32 per batch; all per-batch matrices live in LDS; every matrix
// product is done with fp32-exact V_WMMA_F32_16X16X4_F32 tile MACs.

typedef float v2f __attribute__((ext_vector_type(2)));
typedef float v8f __attribute__((ext_vector_type(8)));

#define WPB 2   // waves (batches) per block

// ---- fragment load/store helpers (ISA 7.12.2 layouts, 32-bit data) -------

// A-matrix 16x4 fragment: lanes 0-15 hold M=0..15/K={0,1}; lanes 16-31 K={2,3}
__device__ __forceinline__ v2f ld_a(const float* m, int ld, int r0, int c0, int lane) {
  const int h = lane >> 4, mr = lane & 15;
  const float* p = m + (r0 + mr) * ld + c0 + (h << 1);
  v2f a; a.x = p[0]; a.y = p[1];
  return a;
}
// B-matrix 4x16 fragment: lanes 0-15 hold N=0..15/K={0,1}; lanes 16-31 K={2,3}
__device__ __forceinline__ v2f ld_b(const float* m, int ld, int r0, int c0, int lane) {
  const int h = lane >> 4, n = lane & 15;
  const float* p = m + (r0 + (h << 1)) * ld + c0 + n;
  v2f b; b.x = p[0]; b.y = p[ld];
  return b;
}
// B fragment gathered from a transposed source: B[k][n] = m[(nc0+n)*ld + kr0+k]
__device__ __forceinline__ v2f ld_bt(const float* m, int ld, int kr0, int nc0, int lane) {
  const int h = lane >> 4, n = lane & 15;
  const float* p = m + (nc0 + n) * ld + kr0 + (h << 1);
  v2f b; b.x = p[0]; b.y = p[1];
  return b;
}

// acc += A[16x16 tile at (ar,ac)] * B[16x16 tile], K=16 via 4x K=4 WMMA.
// BT=true: B tile is the transpose of matrix Bm (tile K-rows br, N-cols bc).
template <bool BT>
__device__ __forceinline__ v8f mma16(v8f acc, const float* A, int lda, int ar, int ac,
                                     const float* Bm, int ldb, int br, int bc, int lane) {
#pragma unroll
  for (int k = 0; k < 16; k += 4) {
    v2f a = ld_a(A, lda, ar, ac + k, lane);
    v2f b = BT ? ld_bt(Bm, ldb, br + k, bc, lane)
               : ld_b (Bm, ldb, br + k, bc, lane);
    acc = __builtin_amdgcn_wmma_f32_16x16x4_f32(false, a, false, b,
                                                (short)0, acc, false, false);
  }
  return acc;
}

// C/D 16x16 store: VGPR r holds M = r + 8*(lane/16), N = lane%16
__device__ __forceinline__ void st_c(float* d, int ld, int r0, int c0, v8f c, int lane) {
  const int h = lane >> 4, n = lane & 15;
#pragma unroll
  for (int r = 0; r < 8; ++r)
    d[(r0 + r + (h << 3)) * ld + c0 + n] = c[r];
}
__device__ __forceinline__ void st_c_add(float* d, int ld, const float* a, int lda,
                                         int r0, int c0, v8f c, int lane) {
  const int h = lane >> 4, n = lane & 15;
#pragma unroll
  for (int r = 0; r < 8; ++r) {
    const int rr = r0 + r + (h << 3);
    d[rr * ld + c0 + n] = c[r] + a[rr * lda + c0 + n];
  }
}
__device__ __forceinline__ void st_c_subfrom(float* d, int ld, int r0, int c0,
                                             v8f c, int lane) {
  const int h = lane >> 4, n = lane & 15;
#pragma unroll
  for (int r = 0; r < 8; ++r) {
    const int idx = (r0 + r + (h << 3)) * ld + c0 + n;
    d[idx] = d[idx] - c[r];
  }
}

__global__ __launch_bounds__(32 * WPB)
void kalman_wmma_kernel(const float* __restrict__ state0, const float* __restrict__ cov0,
                        const float* __restrict__ meas,   const float* __restrict__ Fg,
                        const float* __restrict__ Hg,     const float* __restrict__ Qg,
                        const float* __restrict__ Rg,     float* __restrict__ out,
                        int B, int T) {
  const int tid  = threadIdx.x;
  const int lane = tid & 31;
  const int w    = tid >> 5;

  // Block-shared (read-only) parameter matrices.
  __shared__ float sF[32 * 32];   // 32x32, ld 32
  __shared__ float sH[16 * 32];   // 16x32, ld 32
  __shared__ float sQ[32 * 32];   // 32x32, ld 32
  __shared__ float sR[16 * 16];   // 16x16, ld 16
  // Per-wave scratch.
  __shared__ float sP [WPB][32 * 32];  // covariance / P_pred (ld 32)
  __shared__ float sT1[WPB][32 * 32];  // F*P, then P*H^T (ld 32 / ld 16)
  __shared__ float sT2[WPB][16 * 32];  // H*P_pred (ld 32)
  __shared__ float sSa[WPB][16 * 32];  // [S | I] augmented for inversion (ld 32)
  __shared__ float sKm[WPB][32 * 16];  // Kalman gain (ld 16)
  __shared__ float sx [WPB][32];
  __shared__ float sxp[WPB][32];
  __shared__ float sin_[WPB][16];

  for (int i = tid; i < 32 * 32; i += blockDim.x) sF[i] = Fg[i];
  for (int i = tid; i < 16 * 32; i += blockDim.x) sH[i] = Hg[i];
  for (int i = tid; i < 32 * 32; i += blockDim.x) sQ[i] = Qg[i];
  for (int i = tid; i < 16 * 16; i += blockDim.x) sR[i] = Rg[i];
  __syncthreads();

  const int b = blockIdx.x * WPB + w;
  if (b >= B) return;

  float* P  = sP [w];
  float* T1 = sT1[w];
  float* T2 = sT2[w];
  float* Sa = sSa[w];
  float* Km = sKm[w];
  float* x  = sx [w];
  float* xp = sxp[w];
  float* nu = sin_[w];

  for (int i = lane; i < 32 * 32; i += 32) P[i] = cov0[(size_t)b * 1024 + i];
  x[lane] = state0[(size_t)b * 32 + lane];
  const float* zb = meas + (size_t)b * T * 16;
  float*       ob = out  + (size_t)b * T * 32;
  __builtin_amdgcn_wave_barrier();

#pragma unroll 1
  for (int t = 0; t < T; ++t) {
    if (t + 1 < T) __builtin_prefetch(zb + (size_t)(t + 1) * 16, 0, 1);

    // ---- predict: x_pred = F x ------------------------------------------
    {
      float acc = 0.f;
#pragma unroll
      for (int j = 0; j < 32; ++j) acc += sF[lane * 32 + j] * x[j];
      xp[lane] = acc;
    }
    __builtin_amdgcn_wave_barrier();

    // ---- T1 = F * P ------------------------------------------------------
#pragma unroll
    for (int i = 0; i < 32; i += 16)
#pragma unroll
      for (int j = 0; j < 32; j += 16) {
        v8f acc = {};
        acc = mma16<false>(acc, sF, 32, i, 0,  P, 32, 0,  j, lane);
        acc = mma16<false>(acc, sF, 32, i, 16, P, 32, 16, j, lane);
        st_c(T1, 32, i, j, acc, lane);
      }
    __builtin_amdgcn_wave_barrier();

    // ---- P_pred = T1 * F^T + Q  (into P) --------------------------------
#pragma unroll
    for (int i = 0; i < 32; i += 16)
#pragma unroll
      for (int j = 0; j < 32; j += 16) {
        v8f acc = {};
        acc = mma16<true>(acc, T1, 32, i, 0,  sF, 32, 0,  j, lane);
        acc = mma16<true>(acc, T1, 32, i, 16, sF, 32, 16, j, lane);
        st_c_add(P, 32, sQ, 32, i, j, acc, lane);
      }
    __builtin_amdgcn_wave_barrier();

    // ---- T2 = H * P_pred  (16x32) ---------------------------------------
#pragma unroll
    for (int j = 0; j < 32; j += 16) {
      v8f acc = {};
      acc = mma16<false>(acc, sH, 32, 0, 0,  P, 32, 0,  j, lane);
      acc = mma16<false>(acc, sH, 32, 0, 16, P, 32, 16, j, lane);
      st_c(T2, 32, 0, j, acc, lane);
    }
    __builtin_amdgcn_wave_barrier();

    // ---- Sa = [ T2 * H^T + R | I ] --------------------------------------
    {
      v8f acc = {};
      acc = mma16<true>(acc, T2, 32, 0, 0,  sH, 32, 0,  0, lane);
      acc = mma16<true>(acc, T2, 32, 0, 16, sH, 32, 16, 0, lane);
      st_c_add(Sa, 32, sR, 16, 0, 0, acc, lane);
    }
    for (int i = lane; i < 256; i += 32) {
      const int r = i >> 4, c = i & 15;
      Sa[r * 32 + 16 + c] = (r == c) ? 1.0f : 0.0f;
    }
    __builtin_amdgcn_wave_barrier();

    // ---- innovation: nu = z - H x_pred ----------------------------------
    if (lane < 16) {
      float acc = 0.f;
#pragma unroll
      for (int j = 0; j < 32; ++j) acc += sH[lane * 32 + j] * xp[j];
      nu[lane] = zb[(size_t)t * 16 + lane] - acc;
    }
    __builtin_amdgcn_wave_barrier();

    // ---- invert S in-place (Gauss-Jordan, S is SPD, lane = column) ------
    {
      const int c = lane;
#pragma unroll 1
      for (int k = 0; k < 16; ++k) {
        const float pinv = 1.0f / Sa[k * 32 + k];
        const float rk   = Sa[k * 32 + c] * pinv;
        __builtin_amdgcn_wave_barrier();
        Sa[k * 32 + c] = rk;
        __builtin_amdgcn_wave_barrier();
#pragma unroll 1
        for (int r = 0; r < 16; ++r) {
          if (r == k) continue;
          const float f = Sa[r * 32 + k];
          __builtin_amdgcn_wave_barrier();
          Sa[r * 32 + c] = Sa[r * 32 + c] - f * rk;
          __builtin_amdgcn_wave_barrier();
        }
      }
    }

    // ---- T1 (ld 16) = P_pred * H^T  (32x16) -----------------------------
#pragma unroll
    for (int i = 0; i < 32; i += 16) {
      v8f acc = {};
      acc = mma16<true>(acc, P, 32, i, 0,  sH, 32, 0,  0, lane);
      acc = mma16<true>(acc, P, 32, i, 16, sH, 32, 16, 0, lane);
      st_c(T1, 16, i, 0, acc, lane);
    }
    __builtin_amdgcn_wave_barrier();

    // ---- Km = (P_pred H^T) * S^{-1}  (32x16) ----------------------------
#pragma unroll
    for (int i = 0; i < 32; i += 16) {
      v8f acc = {};
      acc = mma16<false>(acc, T1, 16, i, 0, Sa, 32, 0, 16, lane);
      st_c(Km, 16, i, 0, acc, lane);
    }
    __builtin_amdgcn_wave_barrier();

    // ---- x = x_pred + Km * nu ; emit filtered state ---------------------
    {
      float acc = xp[lane];
#pragma unroll
      for (int j = 0; j < 16; ++j) acc += Km[lane * 16 + j] * nu[j];
      x[lane] = acc;
      ob[(size_t)t * 32 + lane] = acc;
    }
    __builtin_amdgcn_wave_barrier();

    // ---- P = P_pred - Km * (H P_pred) -----------------------------------
#pragma unroll
    for (int i = 0; i < 32; i += 16)
#pragma unroll
      for (int j = 0; j < 32; j += 16) {
        v8f acc = {};
        acc = mma16<false>(acc, Km, 16, i, 0, T2, 32, 0, j, lane);
        st_c_subfrom(P, 32, i, j, acc, lane);
      }
    __builtin_amdgcn_wave_barrier();
  }
}

extern "C" void kernel_launch(void* const* d_in, const int* in_sizes, int n_in,
                              void* d_out, int out_size, void* d_ws, size_t ws_size,
                              hipStream_t stream) {
  const float* state0 = (const float*)d_in[0];
  const float* cov0   = (const float*)d_in[1];
  const float* meas   = (const float*)d_in[2];
  const float* F      = (const float*)d_in[3];
  const float* H      = (const float*)d_in[4];
  const float* Q      = (const float*)d_in[5];
  const float* R      = (const float*)d_in[6];
  float* out = (float*)d_out;

  const int B = in_sizes[0] / 32;
  const int T = in_sizes[2] / (B * 16);
  const int blocks = (B + WPB - 1) / WPB;
  kalman_wmma_kernel<<<blocks, 32 * WPB, 0, stream>>>(state0, cov0, meas,
                                                      F, H, Q, R, out, B, T);
}